// GNN_36593121362163
// MI455X (gfx1250) — compile-verified
//
#include <hip/hip_runtime.h>

typedef float v2f __attribute__((ext_vector_type(2)));
typedef float v8f __attribute__((ext_vector_type(8)));
typedef int   v4i __attribute__((ext_vector_type(4)));

#define N_NODES 100000
#define KC 64      // K-chunk staged in LDS
#define BPAD 72    // padded LDS row stride (floats): 16B-aligned rows, conflict-free halves

// ---------------- gfx1250 async-to-LDS plumbing (guarded) ------------------
// Probe-confirmed signature: (AS1 int4* src, AS3 int4* dst, imm offset, imm cpol)
typedef __attribute__((address_space(1))) v4i g_v4i;
typedef __attribute__((address_space(3))) v4i l_v4i;

#if defined(__gfx1250__) && __has_builtin(__builtin_amdgcn_global_load_async_to_lds_b128)
#define HAVE_ASYNC_LDS 1
#else
#define HAVE_ASYNC_LDS 0
#endif

#if HAVE_ASYNC_LDS
#define ASYNC_CP128(gp, lp) \
  __builtin_amdgcn_global_load_async_to_lds_b128((g_v4i*)(gp), (l_v4i*)(lp), 0, 0)
#if __has_builtin(__builtin_amdgcn_s_wait_asynccnt)
#define WAIT_ASYNC(n) __builtin_amdgcn_s_wait_asynccnt(n)
#else
#define WAIT_ASYNC(n) asm volatile("s_wait_asynccnt %0" ::"i"(n) : "memory")
#endif
#else
#define ASYNC_CP128(gp, lp) (*(float4*)(lp) = *(const float4*)(gp))
#define WAIT_ASYNC(n) ((void)0)
#endif

// ---------- order-preserving float <-> u32 encoding for atomic max ----------
__device__ __forceinline__ unsigned int fenc(float f) {
  unsigned int b = __float_as_uint(f);
  return (b & 0x80000000u) ? ~b : (b | 0x80000000u);
}
__device__ __forceinline__ float fdec(unsigned int u) {
  return __uint_as_float((u & 0x80000000u) ? (u ^ 0x80000000u) : ~u);
}

// ---------------------------------------------------------------------------
// GEMM: out[N,M] = A[N,K] @ W[K,M], fp32 via V_WMMA_F32_16X16X4_F32.
// Block = 4 waves = 64 rows x 64 cols. B panel (KC x 64) double-buffered in
// LDS via async-to-LDS b128; each wave computes a 16x64 tile.
// Fragment layouts per CDNA5 ISA 7.12.2.
// ---------------------------------------------------------------------------
__global__ __launch_bounds__(128) void gemm_wmma_f32(
    const float* __restrict__ A, const float* __restrict__ W,
    float* __restrict__ out, int N, int K, int M) {
  __shared__ float Bs[2][KC * BPAD];   // 2 x 18KB

  const int tid  = threadIdx.x;
  const int wave = tid >> 5;
  const int lane = tid & 31;
  const int half = lane >> 4;
  const int l16  = lane & 15;
  const int r0   = (blockIdx.x * 4 + wave) * 16;
  const int c0   = blockIdx.y * 64;
  const bool active = (r0 < N);

  // staging map: thread t handles (row = t/16 + 8*i, col4 = t&15), i = 0..7
  const int srow = tid >> 4;
  const int scf  = (tid & 15) * 4;

  v8f acc[4] = {};
  const float* Arow =
      active ? (A + (size_t)(r0 + l16) * K + 2 * half) : A;  // safe dummy

  const int nch = K / KC;
  int buf = 0;

  // prologue: stage chunk 0
  {
    const float* g = W + (size_t)srow * M + c0 + scf;
    float* l = &Bs[0][srow * BPAD + scf];
#pragma unroll
    for (int i = 0; i < 8; ++i)
      ASYNC_CP128(g + (size_t)i * 8 * M, l + i * 8 * BPAD);
  }

  for (int c = 0; c < nch; ++c) {
    const int k0c = c * KC;
    if (c + 1 < nch) {  // stage next chunk into other buffer
      const float* g = W + (size_t)(k0c + KC + srow) * M + c0 + scf;
      float* l = &Bs[buf ^ 1][srow * BPAD + scf];
#pragma unroll
      for (int i = 0; i < 8; ++i)
        ASYNC_CP128(g + (size_t)i * 8 * M, l + i * 8 * BPAD);
      WAIT_ASYNC(8);  // in-order: first 8 (chunk c) have landed
      if (active) __builtin_prefetch(Arow + k0c + KC, 0, 0);  // next A chunk
    } else {
      WAIT_ASYNC(0);
    }
    __syncthreads();  // chunk c visible to all waves

    if (active) {
      const float* bbase = &Bs[buf][2 * half * BPAD + l16];
#pragma unroll
      for (int kq = 0; kq < KC; kq += 4) {
        v2f a;
        a.x = Arow[k0c + kq];
        a.y = Arow[k0c + kq + 1];
        const float* bp = bbase + kq * BPAD;
#pragma unroll
        for (int t = 0; t < 4; ++t) {
          v2f b;
          b.x = bp[t * 16];         // row k0+2*half
          b.y = bp[t * 16 + BPAD];  // row k0+2*half+1
          acc[t] = __builtin_amdgcn_wmma_f32_16x16x4_f32(
              false, a, false, b, (short)0, acc[t], false, false);
        }
      }
    }
    __syncthreads();  // everyone done reading buf before it is restaged
    buf ^= 1;
  }

  if (active) {
#pragma unroll
    for (int t = 0; t < 4; ++t) {
#pragma unroll
      for (int g = 0; g < 8; ++g) {
        out[(size_t)(r0 + g + 8 * half) * M + c0 + t * 16 + l16] = acc[t][g];
      }
    }
  }
}

// ---------------------------------------------------------------------------
// Pass A: per-edge attention logits + segment max over dst (u32 atomic max).
// ---------------------------------------------------------------------------
template <int H, int C>
__global__ __launch_bounds__(256) void edge_logits(
    const float* __restrict__ xl, const float* __restrict__ xr,
    const int* __restrict__ ei, const float* __restrict__ att,
    float* __restrict__ logits, unsigned int* __restrict__ mmax,
    int E0, int Etot) {
  constexpr int CT  = H * C;
  constexpr int CPL = CT / 32;
  constexpr int LPH = 32 / H;
  const int wave = threadIdx.x >> 5;
  const int lane = threadIdx.x & 31;
  const int e = blockIdx.x * 8 + wave;
  if (e >= Etot) return;
  const int s = (e < E0) ? ei[e]      : (e - E0);
  const int d = (e < E0) ? ei[E0 + e] : (e - E0);

  const float* pl = xl + (size_t)s * CT + lane * CPL;
  const float* pr = xr + (size_t)d * CT + lane * CPL;
  const float* pa = att + lane * CPL;
  float sum = 0.f;
#pragma unroll
  for (int j = 0; j < CPL; ++j) {
    float v = pl[j] + pr[j];
    v = (v > 0.f) ? v : 0.2f * v;   // GATv2 leaky_relu, slope 0.2
    sum += v * pa[j];
  }
#pragma unroll
  for (int off = 1; off < LPH; off <<= 1) sum += __shfl_xor(sum, off, 32);
  if ((lane & (LPH - 1)) == 0) {
    const int h = lane / LPH;
    logits[(size_t)e * H + h] = sum;
    atomicMax(&mmax[(size_t)d * H + h], fenc(sum));
  }
}

// ---------------------------------------------------------------------------
// Pass B: denom[dst,h] += exp(logit - max[dst,h])
// ---------------------------------------------------------------------------
template <int H>
__global__ __launch_bounds__(256) void edge_denom(
    const float* __restrict__ logits, const unsigned int* __restrict__ mmax,
    const int* __restrict__ ei, float* __restrict__ denom, int E0, int Etot) {
  const int t = blockIdx.x * blockDim.x + threadIdx.x;
  if (t >= Etot * H) return;
  const int e = t / H;
  const int h = t - e * H;
  const int d = (e < E0) ? ei[E0 + e] : (e - E0);
  const float m = fdec(mmax[(size_t)d * H + h]);
  atomicAdd(&denom[(size_t)d * H + h], __expf(logits[t] - m));
}

// ---------------------------------------------------------------------------
// Pass C: out[dst] += alpha * xl[src]
// ---------------------------------------------------------------------------
template <int H, int C>
__global__ __launch_bounds__(256) void edge_scatter(
    const float* __restrict__ xl, const float* __restrict__ logits,
    const unsigned int* __restrict__ mmax, const float* __restrict__ denom,
    const int* __restrict__ ei, float* __restrict__ out, int E0, int Etot) {
  constexpr int CT  = H * C;
  constexpr int CPL = CT / 32;
  constexpr int LPH = 32 / H;
  const int wave = threadIdx.x >> 5;
  const int lane = threadIdx.x & 31;
  const int e = blockIdx.x * 8 + wave;
  if (e >= Etot) return;
  const int s = (e < E0) ? ei[e]      : (e - E0);
  const int d = (e < E0) ? ei[E0 + e] : (e - E0);
  const int h = lane / LPH;
  const float m     = fdec(mmax[(size_t)d * H + h]);
  const float alpha = __expf(logits[(size_t)e * H + h] - m) /
                      denom[(size_t)d * H + h];
  const float* pl = xl  + (size_t)s * CT + lane * CPL;
  float*       po = out + (size_t)d * CT + lane * CPL;
#pragma unroll
  for (int j = 0; j < CPL; ++j) atomicAdd(&po[j], alpha * pl[j]);
}

// ---------------------------------------------------------------------------
__global__ void zero_f32(float* p, int n) {
  int i = blockIdx.x * blockDim.x + threadIdx.x;
  if (i < n) p[i] = 0.f;
}
__global__ void init_u32(unsigned int* p, unsigned int v, int n) {
  int i = blockIdx.x * blockDim.x + threadIdx.x;
  if (i < n) p[i] = v;
}
__global__ void bias_act(const float* __restrict__ in,
                         const float* __restrict__ bias,
                         float* __restrict__ out, int n, int ctMask,
                         float slope) {
  int i = blockIdx.x * blockDim.x + threadIdx.x;
  if (i >= n) return;
  float v = in[i] + bias[i & ctMask];
  out[i] = (v > 0.f) ? v : slope * v;
}
__global__ void masks_k(const float* __restrict__ x, float* __restrict__ om,
                        int n) {
  int i = blockIdx.x * blockDim.x + threadIdx.x;
  if (i >= n) return;
  float v = x[(size_t)i * 128];
  om[i]     = (v == 5.0f) ? 1.0f : 0.0f;  // THIS_AGENT_HE_ID
  om[n + i] = (v == 1.0f) ? 1.0f : 0.0f;  // AGENT_ID
}

// ---------------------------------------------------------------------------
extern "C" void kernel_launch(void* const* d_in, const int* in_sizes, int n_in,
                              void* d_out, int out_size, void* d_ws,
                              size_t ws_size, hipStream_t stream) {
  const float* x    = (const float*)d_in[0];
  const int*   ei   = (const int*)  d_in[1];
  const float* Wl1  = (const float*)d_in[2];
  const float* Wr1  = (const float*)d_in[3];
  const float* att1 = (const float*)d_in[4];
  const float* b1   = (const float*)d_in[5];
  const float* Wl2  = (const float*)d_in[6];
  const float* Wr2  = (const float*)d_in[7];
  const float* att2 = (const float*)d_in[8];
  const float* b2   = (const float*)d_in[9];
  const float* Wl3  = (const float*)d_in[10];
  const float* Wr3  = (const float*)d_in[11];
  const float* att3 = (const float*)d_in[12];
  const float* b3   = (const float*)d_in[13];

  const int N  = N_NODES;
  const int E0 = in_sizes[1] / 2;   // 800000 raw edges
  const int ET = E0 + N;            // + self loops

  float*        bufA   = (float*)d_ws;                             // N*256
  float*        xl     = bufA + (size_t)N * 256;                   // N*256
  float*        xr     = xl   + (size_t)N * 256;                   // N*256
  float*        logits = xr   + (size_t)N * 256;                   // ET*4
  unsigned int* mmax   = (unsigned int*)(logits + (size_t)ET * 4); // N*4
  float*        denom  = (float*)(mmax + (size_t)N * 4);           // N*4

  const int rb = (N / 16 + 3) / 4;
  const dim3 blk128(128), blk256(256);
  const int ewaveGrid = (ET + 7) / 8;

  // ---------------- Layer 1 : 128 -> 4x64, input = x ----------------
  gemm_wmma_f32<<<dim3(rb, 4), blk128, 0, stream>>>(x, Wl1, xl, N, 128, 256);
  gemm_wmma_f32<<<dim3(rb, 4), blk128, 0, stream>>>(x, Wr1, xr, N, 128, 256);
  zero_f32<<<(N * 256 + 255) / 256, blk256, 0, stream>>>(bufA, N * 256);
  init_u32<<<(N * 4 + 255) / 256, blk256, 0, stream>>>(mmax, 0u, N * 4);
  zero_f32<<<(N * 4 + 255) / 256, blk256, 0, stream>>>(denom, N * 4);
  edge_logits<4, 64><<<ewaveGrid, blk256, 0, stream>>>(xl, xr, ei, att1, logits, mmax, E0, ET);
  edge_denom<4><<<(ET * 4 + 255) / 256, blk256, 0, stream>>>(logits, mmax, ei, denom, E0, ET);
  edge_scatter<4, 64><<<ewaveGrid, blk256, 0, stream>>>(xl, logits, mmax, denom, ei, bufA, E0, ET);
  bias_act<<<(N * 256 + 255) / 256, blk256, 0, stream>>>(bufA, b1, bufA, N * 256, 255, 0.01f);

  // ---------------- Layer 2 : 256 -> 4x64 ----------------
  gemm_wmma_f32<<<dim3(rb, 4), blk128, 0, stream>>>(bufA, Wl2, xl, N, 256, 256);
  gemm_wmma_f32<<<dim3(rb, 4), blk128, 0, stream>>>(bufA, Wr2, xr, N, 256, 256);
  zero_f32<<<(N * 256 + 255) / 256, blk256, 0, stream>>>(bufA, N * 256);
  init_u32<<<(N * 4 + 255) / 256, blk256, 0, stream>>>(mmax, 0u, N * 4);
  zero_f32<<<(N * 4 + 255) / 256, blk256, 0, stream>>>(denom, N * 4);
  edge_logits<4, 64><<<ewaveGrid, blk256, 0, stream>>>(xl, xr, ei, att2, logits, mmax, E0, ET);
  edge_denom<4><<<(ET * 4 + 255) / 256, blk256, 0, stream>>>(logits, mmax, ei, denom, E0, ET);
  edge_scatter<4, 64><<<ewaveGrid, blk256, 0, stream>>>(xl, logits, mmax, denom, ei, bufA, E0, ET);
  bias_act<<<(N * 256 + 255) / 256, blk256, 0, stream>>>(bufA, b2, bufA, N * 256, 255, 0.01f);

  // ---------------- Layer 3 : 256 -> 1x128 ----------------
  gemm_wmma_f32<<<dim3(rb, 2), blk128, 0, stream>>>(bufA, Wl3, xl, N, 256, 128);
  gemm_wmma_f32<<<dim3(rb, 2), blk128, 0, stream>>>(bufA, Wr3, xr, N, 256, 128);
  zero_f32<<<(N * 128 + 255) / 256, blk256, 0, stream>>>(bufA, N * 128);
  init_u32<<<(N + 255) / 256, blk256, 0, stream>>>(mmax, 0u, N);
  zero_f32<<<(N + 255) / 256, blk256, 0, stream>>>(denom, N);
  edge_logits<1, 128><<<ewaveGrid, blk256, 0, stream>>>(xl, xr, ei, att3, logits, mmax, E0, ET);
  edge_denom<1><<<(ET + 255) / 256, blk256, 0, stream>>>(logits, mmax, ei, denom, E0, ET);
  edge_scatter<1, 128><<<ewaveGrid, blk256, 0, stream>>>(xl, logits, mmax, denom, ei, bufA, E0, ET);

  float* out_f = (float*)d_out;
  bias_act<<<(N * 128 + 255) / 256, blk256, 0, stream>>>(bufA, b3, out_f, N * 128, 127, 1.0f);
  masks_k<<<(N + 255) / 256, blk256, 0, stream>>>(x, out_f + (size_t)N * 128, N);
}